// BilinearDecoder_14903536517630
// MI455X (gfx1250) — compile-verified
//
#include <hip/hip_runtime.h>
#include <hip/hip_bf16.h>

typedef __attribute__((ext_vector_type(2))) float v2f;
typedef __attribute__((ext_vector_type(8))) float v8f;

#define FEAT 128
#define TILES_PER_BLOCK 8   // 8 x 16 = 128 rows of z per block

// ---------------------------------------------------------------------------
// Kernel 1: V = z @ W   ([n_nodes,128] x [128,128], all f32)
// Uses V_WMMA_F32_16X16X4_F32. One block = 256 threads = 8 wave32s.
// Wave w computes columns [16w, 16w+16) of each 16-row tile.
// B fragments (from W) are loop-invariant per wave -> preloaded to registers.
// A tile staged in LDS with padded stride (132) to avoid bank conflicts.
// Requires n_nodes % 16 == 0 (100000 = 6250*16, holds for this problem).
// ---------------------------------------------------------------------------
__global__ __launch_bounds__(256) void bilinear_gemm_zW(
    const float* __restrict__ z, const float* __restrict__ W,
    float* __restrict__ V, int n_nodes) {
  __shared__ float Al[16][132];   // 16 x 128 A tile, padded (8448 B)

  const int tid   = threadIdx.x;
  const int wv    = tid >> 5;          // wave id 0..7 -> N tile
  const int lane  = tid & 31;
  const int colBase = wv * 16;
  const int mrow  = lane & 15;         // M (for A) / N (for B,D) position
  const int khalf = (lane >> 4) << 1;  // 0 for lanes 0-15, 2 for lanes 16-31

  // Preload loop-invariant B fragments: W[k..k+1][colBase+mrow] for each k-step.
  // Layout per ISA (16x4 f32 mirrored for B 4x16): VGPR0 holds K=k0+khalf,
  // VGPR1 holds K=k0+khalf+1; lanes 0-15 carry K=0,1 and lanes 16-31 K=2,3.
  v2f bf[FEAT / 4];
#pragma unroll
  for (int kk = 0; kk < FEAT / 4; ++kk) {
    const int k = kk * 4 + khalf;
    bf[kk][0] = W[(size_t)k * FEAT + colBase + mrow];
    bf[kk][1] = W[(size_t)(k + 1) * FEAT + colBase + mrow];
  }

  for (int t = 0; t < TILES_PER_BLOCK; ++t) {
    const int rowBase = (blockIdx.x * TILES_PER_BLOCK + t) * 16;
    if (rowBase >= n_nodes) break;          // uniform across block

    __syncthreads();                        // previous tile's Al reads done
    // Cooperative coalesced load of the 16x128 A tile (2048 floats, f4 each).
#pragma unroll
    for (int c4 = tid; c4 < (16 * FEAT) / 4; c4 += 256) {
      const int r = c4 >> 5;                // c4 / (128/4)
      const int c = (c4 & 31) << 2;
      *(float4*)&Al[r][c] = *(const float4*)&z[(size_t)(rowBase + r) * FEAT + c];
    }
    __syncthreads();

    v8f acc = {};
#pragma unroll
    for (int kk = 0; kk < FEAT / 4; ++kk) {
      const int k = kk * 4 + khalf;
      v2f a;
      a[0] = Al[mrow][k];
      a[1] = Al[mrow][k + 1];
      // D = A x B + C  (f32 16x16x4)
      acc = __builtin_amdgcn_wmma_f32_16x16x4_f32(
          false, a, false, bf[kk], (short)0, acc, false, false);
    }

    // Store D: VGPR r -> M = r + (lane>=16 ? 8 : 0), N = mrow (per ISA layout)
    const int mOff = (lane >> 4) << 3;
#pragma unroll
    for (int r = 0; r < 8; ++r) {
      V[(size_t)(rowBase + mOff + r) * FEAT + colBase + mrow] = acc[r];
    }
  }
}

// ---------------------------------------------------------------------------
// Kernel 2: score[e] = dot(V[src[e]], z[dst[e]])
// One wave32 per edge iteration: 32 lanes x float4 = one coalesced 512B row
// from each table (both tables ~51MB -> resident in the 192MB L2), then a
// 5-step shfl_xor wave reduction.
// ---------------------------------------------------------------------------
__global__ __launch_bounds__(256) void bilinear_edge_scores(
    const float* __restrict__ V, const float* __restrict__ z,
    const int* __restrict__ e_src, const int* __restrict__ e_dst,
    float* __restrict__ out, int n_edges) {
  const int lane   = threadIdx.x & 31;
  const int wid    = (blockIdx.x * blockDim.x + threadIdx.x) >> 5;
  const int nwaves = (gridDim.x * blockDim.x) >> 5;

  for (int e = wid; e < n_edges; e += nwaves) {
    const int s = e_src[e];
    const int d = e_dst[e];
    const float4 av = ((const float4*)(V + (size_t)s * FEAT))[lane];
    const float4 bv = ((const float4*)(z + (size_t)d * FEAT))[lane];
    float sum = av.x * bv.x + av.y * bv.y + av.z * bv.z + av.w * bv.w;
#pragma unroll
    for (int off = 16; off > 0; off >>= 1) sum += __shfl_xor(sum, off, 32);
    if (lane == 0) out[e] = sum;
  }
}

extern "C" void kernel_launch(void* const* d_in, const int* in_sizes, int n_in,
                              void* d_out, int out_size, void* d_ws, size_t ws_size,
                              hipStream_t stream) {
  const float* z  = (const float*)d_in[0];         // [n_nodes, 128]
  const int*   ei = (const int*)d_in[1];           // [2, n_edges]
  const float* W  = (const float*)d_in[2];         // [128, 128]
  float* out = (float*)d_out;                      // [n_edges]
  float* V   = (float*)d_ws;                       // [n_nodes, 128] scratch

  const int n_nodes = in_sizes[0] / FEAT;          // 100000
  const int n_edges = in_sizes[1] / 2;             // 600000

  const int rowTiles = (n_nodes + 15) / 16;
  const int gemmBlocks = (rowTiles + TILES_PER_BLOCK - 1) / TILES_PER_BLOCK;
  bilinear_gemm_zW<<<gemmBlocks, 256, 0, stream>>>(z, W, V, n_nodes);

  bilinear_edge_scores<<<2048, 256, 0, stream>>>(V, z, ei, ei + n_edges, out,
                                                 n_edges);
}